// NavalModel_15556371546375
// MI455X (gfx1250) — compile-verified
//
#include <hip/hip_runtime.h>
#include <hip/hip_bf16.h>

#define NF 64
#define NC 16
#define LEN 2048
#define BATCH 256
#define BETA 2.5f
#define KAPPA 10.0f
#define EPSN 1e-12f
#define W_BI 0.01f
#define W_L2 0.01f

typedef float v2f __attribute__((ext_vector_type(2)));
typedef float v8f __attribute__((ext_vector_type(8)));
typedef unsigned int v4u __attribute__((ext_vector_type(4)));
typedef int v4i __attribute__((ext_vector_type(4)));
typedef int v8i __attribute__((ext_vector_type(8)));

__device__ __forceinline__ float clip01(float v) { return fminf(fmaxf(v, 0.f), 1.f); }
__device__ __forceinline__ float fast_rcp(float v) { return __builtin_amdgcn_rcpf(v); }

// ---------------------------------------------------------------------------
// Kernel 0: temporal window table wtab[f][l] = sig(B*(l-t1u))*sig(B*(t2u-l)) + eps
// ---------------------------------------------------------------------------
__global__ void build_wtab(const float* __restrict__ t1, const float* __restrict__ t2,
                           float* __restrict__ wtab) {
    int idx = blockIdx.x * blockDim.x + threadIdx.x;   // 0 .. NF*LEN-1
    int f = idx >> 11;
    int l = idx & (LEN - 1);
    float t1c = fminf(fmaxf(t1[f], 0.f), (float)(LEN - 1));
    float t2c = fminf(fmaxf(t2[f], 0.f), (float)(LEN - 1));
    if (t1c >= t2c - 1.f) t1c = t2c - 1.f;
    float lf = (float)l;
    float s1 = fast_rcp(1.f + __expf(-BETA * (lf - t1c)));
    float s2 = fast_rcp(1.f + __expf(-BETA * (t2c - lf)));
    wtab[idx] = s1 * s2 + EPSN;
}

// ---------------------------------------------------------------------------
// Kernel 1: r1[b][f]. One block per batch element.
// x[b] (16KB contiguous) is DMA'd into LDS by wave 0 via the Tensor Data
// Mover (tensor_load_to_lds + s_wait_tensorcnt), then 8 waves each own 4
// feature PAIRS (even dim0 / odd dim1) and sweep L two timesteps at a time:
// one ds_load_b128 + two global_load_b64 feed 8 v_exp_f32.
// ---------------------------------------------------------------------------
__global__ void stage1(const float* __restrict__ x, const float* __restrict__ a,
                       const float* __restrict__ b, const float* __restrict__ tvt,
                       const float* __restrict__ wtab, float* __restrict__ r1g) {
    __shared__ __align__(16) float xs[LEN * 2];
    const int bi = blockIdx.x;

    // ---- TDM: wave 0 issues one descriptor load of 4096 f32 -> LDS ----
    if (threadIdx.x < 32) {
        unsigned lds_addr = (unsigned)(uintptr_t)(&xs[0]);
        unsigned long long ga =
            (unsigned long long)(uintptr_t)(x + (size_t)bi * LEN * 2);
        v4u g0;
        g0.x = 1u;                                           // count=1 (valid), user D#
        g0.y = lds_addr;                                     // lds_addr [63:32]
        g0.z = (unsigned)(ga & 0xFFFFFFFFu);                 // global_addr lo
        g0.w = (unsigned)((ga >> 32) & 0x01FFFFFFu) | (2u << 30); // addr hi | type=2
        v8i g1;
        g1[0] = (int)(2u << 16);        // wg_mask=0, data_size=2 (4B)
        g1[1] = (int)(4096u << 16);     // tensor_dim0[15:0]=4096 at bits[63:48]
        g1[2] = (int)(1u << 16);        // tensor_dim0 hi=0, tensor_dim1=1
        g1[3] = (int)(4096u << 16);     // tensor_dim1 hi=0, tile_dim0=4096
        g1[4] = 1;                       // tile_dim1=1, tile_dim2=0
        g1[5] = 4096;                    // tensor_dim0_stride lo
        g1[6] = 0;
        g1[7] = 0;
        v4i gz = {0, 0, 0, 0};
#if __clang_major__ >= 23
        v8i gz8 = {0, 0, 0, 0, 0, 0, 0, 0};
        __builtin_amdgcn_tensor_load_to_lds(g0, g1, gz, gz, gz8, 0);
#else
        __builtin_amdgcn_tensor_load_to_lds(g0, g1, gz, gz, 0);
#endif
        __builtin_amdgcn_s_wait_tensorcnt(0);
    }
    __syncthreads();

    const int wave = threadIdx.x >> 5;
    const int lane = threadIdx.x & 31;
    const float4* __restrict__ xv4 = (const float4*)xs;

    for (int fp = 0; fp < 4; ++fp) {
        const int f0 = wave * 8 + fp * 2;      // even feature -> x dim 0
        const int f1 = f0 + 1;                 // odd  feature -> x dim 1
        const float a0 = a[f0], b0 = b[f0];
        const float a1 = a[f1], b1 = b[f1];
        const float2* __restrict__ w0row = (const float2*)(wtab + f0 * LEN);
        const float2* __restrict__ w1row = (const float2*)(wtab + f1 * LEN);

        float dp0 = 0.f, np0 = 0.f, dm0 = 0.f, nm0 = 0.f;
        float dp1 = 0.f, np1 = 0.f, dm1 = 0.f, nm1 = 0.f;

        for (int i = lane; i < LEN / 2; i += 32) {
            float4 xv = xv4[i];               // {x[2i,0], x[2i,1], x[2i+1,0], x[2i+1,1]}
            float2 w0 = w0row[i];             // w+eps for f0 at l=2i,2i+1
            float2 w1 = w1row[i];
            // timestep 2i
            float rpa = fmaf(a0, xv.x, -b0);
            float rpb = fmaf(a1, xv.y, -b1);
            // timestep 2i+1
            float rpc = fmaf(a0, xv.z, -b0);
            float rpd = fmaf(a1, xv.w, -b1);
            float eap = __expf(BETA * rpa), eam = __expf(-BETA * rpa);
            float ebp = __expf(BETA * rpb), ebm = __expf(-BETA * rpb);
            float ecp = __expf(BETA * rpc), ecm = __expf(-BETA * rpc);
            float edp = __expf(BETA * rpd), edm = __expf(-BETA * rpd);
            float wap = w0.x * eap, wcp = w0.y * ecp;
            float wam = w0.x * eam, wcm = w0.y * ecm;
            float wbp = w1.x * ebp, wdp = w1.y * edp;
            float wbm = w1.x * ebm, wdm = w1.y * edm;
            dp0 += wap + wcp;  np0 = fmaf(wap, rpa, fmaf(wcp, rpc, np0));
            dm0 += wam + wcm;  nm0 = fmaf(wam, rpa, fmaf(wcm, rpc, nm0));
            dp1 += wbp + wdp;  np1 = fmaf(wbp, rpb, fmaf(wdp, rpd, np1));
            dm1 += wbm + wdm;  nm1 = fmaf(wbm, rpb, fmaf(wdm, rpd, nm1));
        }
        // wave32 butterfly reduction of 8 accumulators
        #pragma unroll
        for (int off = 16; off > 0; off >>= 1) {
            dp0 += __shfl_xor(dp0, off, 32);  np0 += __shfl_xor(np0, off, 32);
            dm0 += __shfl_xor(dm0, off, 32);  nm0 += __shfl_xor(nm0, off, 32);
            dp1 += __shfl_xor(dp1, off, 32);  np1 += __shfl_xor(np1, off, 32);
            dm1 += __shfl_xor(dm1, off, 32);  nm1 += __shfl_xor(nm1, off, 32);
        }
        if (lane == 0) {
            float tv0 = clip01(tvt[f0]);
            float tv1 = clip01(tvt[f1]);
            float r10 = tv0 * (np0 * fast_rcp(dp0)) + (1.f - tv0) * (nm0 * fast_rcp(dm0));
            float r11 = tv1 * (np1 * fast_rcp(dp1)) + (1.f - tv1) * (nm1 * fast_rcp(dm1));
            r1g[bi * NF + f0] = r10;
            r1g[bi * NF + f1] = r11;
        }
    }
}

// ---------------------------------------------------------------------------
// Kernel 2: stages 2+3 via WMMA. One wave handles 16 batch columns.
//   D(16c x 16b) = P1eps(16 x 64) * E(64 x 16b),  K=64 as 16 chained
//   V_WMMA_F32_16X16X4_F32, four accumulators (or/and x num/den).
// ---------------------------------------------------------------------------
__global__ void stage23(const float* __restrict__ r1g, const float* __restrict__ p1,
                        const float* __restrict__ p2, const float* __restrict__ tvl1,
                        const float* __restrict__ tvl2, float* __restrict__ out) {
    const int lane = threadIdx.x;       // 0..31, EXEC all ones before WMMAs
    const int col  = lane & 15;
    const int half = lane >> 4;
    const int bcol = blockIdx.x * 16 + col;

    v8f Dor_den = {}, Dor_num = {}, Dand_den = {}, Dand_num = {};

    #pragma unroll 4
    for (int kc = 0; kc < 16; ++kc) {
        const int fbase = kc * 4 + half * 2;
        // A operand: clipped p1 + eps (row m = col, K pair)
        float2 pv = *(const float2*)(p1 + col * NF + fbase);
        v2f A;
        A.x = clip01(pv.x) + EPSN;
        A.y = clip01(pv.y) + EPSN;
        // B operands: exp factors of r1 (column n = col -> batch bcol)
        float2 rv = *(const float2*)(r1g + bcol * NF + fbase);
        float e0p = __expf(KAPPA * rv.x),  e1p = __expf(KAPPA * rv.y);
        float e0m = __expf(-KAPPA * rv.x), e1m = __expf(-KAPPA * rv.y);
        v2f Bdp = {e0p, e1p};
        v2f Bnp = {e0p * rv.x, e1p * rv.y};
        v2f Bdm = {e0m, e1m};
        v2f Bnm = {e0m * rv.x, e1m * rv.y};

        Dor_den  = __builtin_amdgcn_wmma_f32_16x16x4_f32(false, A, false, Bdp, (short)0, Dor_den,  false, false);
        Dor_num  = __builtin_amdgcn_wmma_f32_16x16x4_f32(false, A, false, Bnp, (short)0, Dor_num,  false, false);
        Dand_den = __builtin_amdgcn_wmma_f32_16x16x4_f32(false, A, false, Bdm, (short)0, Dand_den, false, false);
        Dand_num = __builtin_amdgcn_wmma_f32_16x16x4_f32(false, A, false, Bnm, (short)0, Dand_num, false, false);
    }

    // Stage 3: lane holds (c = v + 8*half, b = bcol) in D VGPR v.
    const float tl2 = clip01(tvl2[0]);
    float s_np = 0.f, s_dp = 0.f, s_nm = 0.f, s_dm = 0.f;
    #pragma unroll
    for (int v = 0; v < 8; ++v) {
        int c = v + half * 8;
        float r_or  = Dor_num[v]  * fast_rcp(Dor_den[v]);
        float r_and = Dand_num[v] * fast_rcp(Dand_den[v]);
        float tl = clip01(tvl1[c]);
        float r2 = tl * r_or + (1.f - tl) * r_and;
        float p2c = clip01(p2[c]) + EPSN;
        float e2p = __expf(KAPPA * r2);
        float e2m = __expf(-KAPPA * r2);
        s_dp = fmaf(p2c, e2p, s_dp);
        s_np = fmaf(p2c * e2p, r2, s_np);
        s_dm = fmaf(p2c, e2m, s_dm);
        s_nm = fmaf(p2c * e2m, r2, s_nm);
    }
    // combine the two lane halves (c=0..7 with c=8..15)
    s_dp += __shfl_xor(s_dp, 16, 32);
    s_np += __shfl_xor(s_np, 16, 32);
    s_dm += __shfl_xor(s_dm, 16, 32);
    s_nm += __shfl_xor(s_nm, 16, 32);

    if (half == 0) {
        float r_or2  = s_np * fast_rcp(s_dp);
        float r_and2 = s_nm * fast_rcp(s_dm);
        out[bcol] = tl2 * r_or2 + (1.f - tl2) * r_and2;
    }
}

// ---------------------------------------------------------------------------
// Kernel 3: regularizer -> out[BATCH]
// ---------------------------------------------------------------------------
__global__ void regk(const float* __restrict__ p1, const float* __restrict__ p2,
                     const float* __restrict__ tvt, const float* __restrict__ tvl1,
                     const float* __restrict__ tvl2, float* __restrict__ out) {
    __shared__ float red[256];
    const int t = threadIdx.x;
    float acc = 0.f;
    for (int i = t; i < NC * NF; i += 256) {
        float p = clip01(p1[i]);
        acc += W_BI * p * (1.f - p) + W_L2 * p * p;
    }
    if (t < NC) {
        float p = clip01(p2[t]);
        acc += W_BI * p * (1.f - p) + W_L2 * p * p;
        float tl = clip01(tvl1[t]);
        acc += W_BI * tl * (1.f - tl);
    }
    if (t < NF) {
        float tv = clip01(tvt[t]);
        acc += W_BI * tv * (1.f - tv);
    }
    if (t == 0) {
        float tl2 = clip01(tvl2[0]);
        acc += W_BI * tl2 * (1.f - tl2);
    }
    red[t] = acc;
    __syncthreads();
    for (int s = 128; s > 0; s >>= 1) {
        if (t < s) red[t] += red[t + s];
        __syncthreads();
    }
    if (t == 0) out[BATCH] = red[0];
}

// ---------------------------------------------------------------------------
// Entry point. Inputs (all f32): x,t1,t2,a,b,p1,p2,tvt,tvl1,tvl2,eps(unused).
// Out: 257 floats (r[256], reg). Workspace: wtab 512KB + r1 64KB = 576KB.
// ---------------------------------------------------------------------------
extern "C" void kernel_launch(void* const* d_in, const int* in_sizes, int n_in,
                              void* d_out, int out_size, void* d_ws, size_t ws_size,
                              hipStream_t stream) {
    const float* x    = (const float*)d_in[0];
    const float* t1   = (const float*)d_in[1];
    const float* t2   = (const float*)d_in[2];
    const float* a    = (const float*)d_in[3];
    const float* b    = (const float*)d_in[4];
    const float* p1   = (const float*)d_in[5];
    const float* p2   = (const float*)d_in[6];
    const float* tvt  = (const float*)d_in[7];
    const float* tvl1 = (const float*)d_in[8];
    const float* tvl2 = (const float*)d_in[9];
    float* out  = (float*)d_out;
    float* wtab = (float*)d_ws;                 // NF*LEN floats
    float* r1g  = wtab + NF * LEN;              // BATCH*NF floats

    build_wtab<<<(NF * LEN) / 256, 256, 0, stream>>>(t1, t2, wtab);
    stage1<<<BATCH, 256, 0, stream>>>(x, a, b, tvt, wtab, r1g);
    stage23<<<BATCH / 16, 32, 0, stream>>>(r1g, p1, p2, tvl1, tvl2, out);
    regk<<<1, 256, 0, stream>>>(p1, p2, tvt, tvl1, tvl2, out);
}